// PositionAwarePooling_21715354649822
// MI455X (gfx1250) — compile-verified
//
#include <hip/hip_runtime.h>

typedef float f4 __attribute__((ext_vector_type(4)));

// x: [32, 128, 128, 128] fp32, out: [32, 384, 64, 64] fp32
// One thread: one (b,c,oh) row-pair segment of 8 input columns -> 4 outputs/plane.
// Loads: 4 x global_load_b128 (NT), Stores: 3 x global_store_b128 (NT).
__global__ __launch_bounds__(256) void pap_pool_kernel(const float* __restrict__ x,
                                                       float* __restrict__ out,
                                                       unsigned total_groups) {
    unsigned gid = blockIdx.x * blockDim.x + threadIdx.x;
    if (gid >= total_groups) return;

    // Decompose: gid -> (b, c, oh, q) ; q = which group of 4 output cols (16 per row)
    unsigned q  = gid & 15u;          // 0..15  (4 output cols each -> 64 cols/row)
    unsigned r  = gid >> 4;
    unsigned oh = r & 63u;            // pooled row 0..63
    r >>= 6;
    unsigned c  = r & 127u;           // channel 0..127
    unsigned b  = r >> 7;             // batch 0..31

    // Input: rows ih0 = 2*oh, ih0+1 ; cols [q*8, q*8+8)
    const size_t plane_in = 128u * 128u;                       // H*W
    const float* src = x + ((size_t)(b * 128u + c)) * plane_in
                         + (size_t)(oh * 2u) * 128u + q * 8u;

    f4 top0 = __builtin_nontemporal_load((const f4*)(src));         // row ih0, cols 0..3
    f4 top1 = __builtin_nontemporal_load((const f4*)(src + 4));     // row ih0, cols 4..7
    f4 bot0 = __builtin_nontemporal_load((const f4*)(src + 128));   // row ih0+1, cols 0..3
    f4 bot1 = __builtin_nontemporal_load((const f4*)(src + 132));   // row ih0+1, cols 4..7

    f4 top[2] = { top0, top1 };
    f4 bot[2] = { bot0, bot1 };

    const float inv128 = 1.0f / 128.0f;
    f4 vals, ph, pw;

    #pragma unroll
    for (int i = 0; i < 4; ++i) {
        // window i covers input cols (2i, 2i+1) within this 8-col segment
        float p0 = top[i >> 1][(i & 1) * 2];       // (0,0) -> idx 0
        float p1 = top[i >> 1][(i & 1) * 2 + 1];   // (0,1) -> idx 1
        float p2 = bot[i >> 1][(i & 1) * 2];       // (1,0) -> idx 2
        float p3 = bot[i >> 1][(i & 1) * 2 + 1];   // (1,1) -> idx 3

        // first-max tie-break (strict >) == jnp.argmax semantics
        float v = p0; int idx = 0;
        if (p1 > v) { v = p1; idx = 1; }
        if (p2 > v) { v = p2; idx = 2; }
        if (p3 > v) { v = p3; idx = 3; }

        unsigned ow  = q * 4u + (unsigned)i;
        int row = (int)(oh * 2u) + (idx >> 1);
        int col = (int)(ow * 2u) + (idx & 1);

        vals[i] = v;
        ph[i]   = (float)row * inv128;   // row / H
        pw[i]   = (float)col * inv128;   // col / W
    }

    // out[b][plane][oh][ow] ; planes: vals=c, pos_h=c+128, pos_w=c+256
    const size_t plane_out = 64u * 64u;                           // h*w
    size_t obase = (((size_t)b * 384u + c) * 64u + oh) * 64u + q * 4u;

    __builtin_nontemporal_store(vals, (f4*)(out + obase));
    __builtin_nontemporal_store(ph,   (f4*)(out + obase + 128u * plane_out));
    __builtin_nontemporal_store(pw,   (f4*)(out + obase + 256u * plane_out));
}

extern "C" void kernel_launch(void* const* d_in, const int* in_sizes, int n_in,
                              void* d_out, int out_size, void* d_ws, size_t ws_size,
                              hipStream_t stream) {
    const float* x = (const float*)d_in[0];
    float* out = (float*)d_out;

    // Each thread consumes 16 input elements (8 cols x 2 rows).
    unsigned total_groups = (unsigned)(in_sizes[0] / 16);   // 32*128*64*16 = 4,194,304
    unsigned block = 256;
    unsigned grid = (total_groups + block - 1) / block;     // 16384

    pap_pool_kernel<<<grid, block, 0, stream>>>(x, out, total_groups);
}